// HeterGConv_Edge_56324201120496
// MI455X (gfx1250) — compile-verified
//
#include <hip/hip_runtime.h>

// Problem constants (match reference)
#define L_DIM 2000
#define NN    (3 * L_DIM)     // 6000 nodes
#define F_DIM 512

// GEMM tiling: 128x64 block tile, K-block 32, 8 waves (wave32) per block,
// each wave computes a 32x32 micro-tile (4 WMMA accumulators).
#define TM 128
#define TN 64
#define TK 32
#define LDP 36                 // padded LDS row stride (floats): 16B-aligned f4
                               // stores, conflict-free 16-lane fragment reads

typedef __attribute__((ext_vector_type(2))) float v2f;
typedef __attribute__((ext_vector_type(8))) float v8f;

// ---------------------------------------------------------------- utilities
__global__ void zero_kernel(float* __restrict__ p, int n) {
    int i = blockIdx.x * blockDim.x + threadIdx.x;
    if (i < n) p[i] = 0.0f;
}

// row_sum[r] += w  for each edge (r, c)
__global__ void degree_kernel(const int* __restrict__ ei,
                              const float* __restrict__ ew,
                              float* __restrict__ rowsum, int E) {
    int e = blockIdx.x * blockDim.x + threadIdx.x;
    if (e < E) atomicAdd(&rowsum[ei[e]], ew[e]);
}

// in-place: rowsum -> d_inv_sqrt (0 where row_sum <= 0)
__global__ void dinv_kernel(float* __restrict__ rs, int n) {
    int i = blockIdx.x * blockDim.x + threadIdx.x;
    if (i < n) {
        float s = rs[i];
        rs[i] = (s > 0.0f) ? rsqrtf(s) : 0.0f;
    }
}

// feature = concat(x_a, x_v, x_t); feature_sum (d_out) = feature
__global__ void concat_init_kernel(const float* __restrict__ xa,
                                   const float* __restrict__ xv,
                                   const float* __restrict__ xt,
                                   float* __restrict__ feat,
                                   float* __restrict__ out) {
    int i = blockIdx.x * blockDim.x + threadIdx.x;
    const int seg = L_DIM * F_DIM;
    if (i < NN * F_DIM) {
        int part = i / seg;
        int rem  = i - part * seg;
        const float* src = (part == 0) ? xa : ((part == 1) ? xv : xt);
        float v = src[rem];
        feat[i] = v;
        out[i]  = v;
    }
}

// Z[r][:] += w_e * H[c][:]   (H already scaled by d_inv_sqrt[c] in GEMM epilogue)
// One thread per (edge, float4-chunk). H and Z are L2-resident (12.3 MB each).
__global__ void scatter_kernel(const int* __restrict__ ei,
                               const float* __restrict__ ew,
                               const float* __restrict__ H,
                               float* __restrict__ Z, int E) {
    long long g = (long long)blockIdx.x * blockDim.x + threadIdx.x;
    const int C4 = F_DIM / 4; // 128
    if (g >= (long long)E * C4) return;
    int e  = (int)(g >> 7);
    int c4 = (int)(g & 127);
    int r = ei[e];
    int c = ei[E + e];
    float w = ew[e];
    float4 h = ((const float4*)(H + (size_t)c * F_DIM))[c4];
    float* z = Z + (size_t)r * F_DIM + c4 * 4;
    atomicAdd(z + 0, w * h.x);
    atomicAdd(z + 1, w * h.y);
    atomicAdd(z + 2, w * h.z);
    atomicAdd(z + 3, w * h.w);
}

// feature = relu(d_inv_sqrt[r] * Z + gcn_b)
__global__ void gcn_post_kernel(const float* __restrict__ Z,
                                const float* __restrict__ dinv,
                                const float* __restrict__ bias,
                                float* __restrict__ feat) {
    int i = blockIdx.x * blockDim.x + threadIdx.x;
    if (i < NN * F_DIM) {
        int m = i >> 9;       // / 512
        int n = i & 511;
        float t = dinv[m] * Z[i] + bias[n];
        feat[i] = t > 0.0f ? t : 0.0f;
    }
}

// ------------------------------------------------------------- WMMA fp32 GEMM
// C(M x 512) = X(M x 512) @ W(512 x 512) via V_WMMA_F32_16X16X4_F32.
// mode 1: Hout[m][n] = acc * rowscale[m]            (gcn branch, d[c] folding)
// mode 2: Acc[m][n] += leaky_relu(acc + bias[n])    (fc branch, feature_sum)
__global__ __launch_bounds__(256)
void gemm_wmma_kernel(const float* __restrict__ X,
                      const float* __restrict__ W,
                      float* __restrict__ Hout,
                      float* __restrict__ Acc,
                      const float* __restrict__ rowscale,
                      const float* __restrict__ bias,
                      int M, int mode) {
    const int K = F_DIM, N = F_DIM;
    __shared__ float As[TM][LDP];   // A tile, row-major [m][k]   (18 KB)
    __shared__ float Bs[TN][LDP];   // B tile, TRANSPOSED [n][k]  ( 9 KB)

    const int tid    = threadIdx.x;
    const int lane   = tid & 31;
    const int waveId = tid >> 5;   // 0..7
    const int wy = waveId >> 1;    // 0..3 -> 32-row strip
    const int wx = waveId & 1;     // 0..1 -> 32-col strip
    const int m0 = wy * 32;
    const int n0 = wx * 32;
    const int mBase = blockIdx.y * TM;
    const int nBase = blockIdx.x * TN;

    // ISA 16x16x4 f32 fragment mapping (wave32):
    // A: lane holds A[m = lane&15][kb..kb+1], kb = 2*(lane>>4)
    // B: lane holds B[kb..kb+1][n = lane&15]  -> contiguous in Bs[n][k]
    const int fr = lane & 15;
    const int kb = (lane >> 4) << 1;

    v8f c00 = {}, c01 = {}, c10 = {}, c11 = {};

    for (int k0 = 0; k0 < K; k0 += TK) {
        // Stage A tile: 128x32 floats = 1024 float4, 4 per thread.
#pragma unroll
        for (int it = 0; it < 4; ++it) {
            int i4 = tid + it * 256;            // 0..1023
            int ar = i4 >> 3, ac = (i4 & 7) << 2;
            int gm = mBase + ar;
            float4 av = make_float4(0.f, 0.f, 0.f, 0.f);
            if (gm < M) av = *(const float4*)&X[(size_t)gm * K + k0 + ac];
            *(float4*)&As[ar][ac] = av;         // (ar*LDP+ac)*4 is 16B aligned
        }
        // Stage B tile transposed: 32x64 floats = 512 float4, 2 per thread.
#pragma unroll
        for (int it = 0; it < 2; ++it) {
            int i4 = tid + it * 256;            // 0..511
            int br = i4 >> 4, bc = (i4 & 15) << 2;
            float4 bv = *(const float4*)&W[(size_t)(k0 + br) * N + nBase + bc];
            Bs[bc + 0][br] = bv.x;
            Bs[bc + 1][br] = bv.y;
            Bs[bc + 2][br] = bv.z;
            Bs[bc + 3][br] = bv.w;
        }
        __syncthreads();

        // Prefetch next K tile into cache (global_prefetch_b8).
        if (k0 + TK < K) {
            int gm = mBase + (tid & 127);
            if (gm < M) __builtin_prefetch(&X[(size_t)gm * K + k0 + TK], 0, 0);
            __builtin_prefetch(&W[(size_t)(k0 + TK + (tid & 31)) * N + nBase], 0, 0);
        }

#pragma unroll
        for (int kk = 0; kk < TK; kk += 4) {
            v2f a0 = *(const v2f*)&As[m0 + fr][kk + kb];
            v2f a1 = *(const v2f*)&As[m0 + 16 + fr][kk + kb];
            v2f b0 = *(const v2f*)&Bs[n0 + fr][kk + kb];
            v2f b1 = *(const v2f*)&Bs[n0 + 16 + fr][kk + kb];
            // (neg_a, A, neg_b, B, c_mod, C, reuse_a, reuse_b)
            c00 = __builtin_amdgcn_wmma_f32_16x16x4_f32(
                      false, a0, false, b0, (short)0, c00, false, false);
            c01 = __builtin_amdgcn_wmma_f32_16x16x4_f32(
                      false, a0, false, b1, (short)0, c01, false, false);
            c10 = __builtin_amdgcn_wmma_f32_16x16x4_f32(
                      false, a1, false, b0, (short)0, c10, false, false);
            c11 = __builtin_amdgcn_wmma_f32_16x16x4_f32(
                      false, a1, false, b1, (short)0, c11, false, false);
        }
        __syncthreads();
    }

    // Epilogue. C/D layout: VGPR r: lanes 0-15 -> M=r, lanes 16-31 -> M=r+8;
    // N = lane&15.
    const int nc = lane & 15;
    const int mh = (lane >> 4) << 3;   // 0 or 8
#pragma unroll
    for (int r = 0; r < 8; ++r) {
        int mA = mBase + m0 + mh + r;        // rows of c00/c01
        int mB = mA + 16;                    // rows of c10/c11
        int nA = nBase + n0 + nc;
        int nB = nA + 16;
        if (mode == 1) {
            if (mA < M) {
                float s = rowscale[mA];
                Hout[(size_t)mA * N + nA] = c00[r] * s;
                Hout[(size_t)mA * N + nB] = c01[r] * s;
            }
            if (mB < M) {
                float s = rowscale[mB];
                Hout[(size_t)mB * N + nA] = c10[r] * s;
                Hout[(size_t)mB * N + nB] = c11[r] * s;
            }
        } else {
            if (mA < M) {
                float t0 = c00[r] + bias[nA];
                float t1 = c01[r] + bias[nB];
                t0 = t0 > 0.0f ? t0 : 0.01f * t0;
                t1 = t1 > 0.0f ? t1 : 0.01f * t1;
                Acc[(size_t)mA * N + nA] += t0;
                Acc[(size_t)mA * N + nB] += t1;
            }
            if (mB < M) {
                float t0 = c10[r] + bias[nA];
                float t1 = c11[r] + bias[nB];
                t0 = t0 > 0.0f ? t0 : 0.01f * t0;
                t1 = t1 > 0.0f ? t1 : 0.01f * t1;
                Acc[(size_t)mB * N + nA] += t0;
                Acc[(size_t)mB * N + nB] += t1;
            }
        }
    }
}

// -------------------------------------------------------------------- driver
extern "C" void kernel_launch(void* const* d_in, const int* in_sizes, int n_in,
                              void* d_out, int out_size, void* d_ws, size_t ws_size,
                              hipStream_t stream) {
    const float* xa   = (const float*)d_in[0];
    const float* xv   = (const float*)d_in[1];
    const float* xt   = (const float*)d_in[2];
    const float* ewp  = (const float*)d_in[3];   // edge_weight_param (first E used)
    const float* gcnW = (const float*)d_in[4];   // [2][512][512]
    const float* gcnB = (const float*)d_in[5];   // [2][512]
    const float* fcW  = (const float*)d_in[6];   // [2][512][512]
    const float* fcB  = (const float*)d_in[7];   // [2][512]
    const int*   ei   = (const int*)d_in[8];     // [2][E]
    const int E = in_sizes[8] / 2;

    // Workspace layout: feat | H | Z | dinv   (~37 MB total)
    float* feat = (float*)d_ws;
    float* H    = feat + (size_t)NN * F_DIM;
    float* Z    = H    + (size_t)NN * F_DIM;
    float* dinv = Z    + (size_t)NN * F_DIM;
    float* out  = (float*)d_out;

    const int tot = NN * F_DIM;

    // degree -> d_inv_sqrt
    zero_kernel<<<(NN + 255) / 256, 256, 0, stream>>>(dinv, NN);
    degree_kernel<<<(E + 255) / 256, 256, 0, stream>>>(ei, ewp, dinv, E);
    dinv_kernel<<<(NN + 255) / 256, 256, 0, stream>>>(dinv, NN);

    // feature = concat; feature_sum (d_out) = feature
    concat_init_kernel<<<(tot + 255) / 256, 256, 0, stream>>>(xa, xv, xt, feat, out);

    dim3 ggrid(F_DIM / TN, (NN + TM - 1) / TM);   // 8 x 47
    long long sc_tot = (long long)E * (F_DIM / 4);
    int sc_blocks = (int)((sc_tot + 255) / 256);

    for (int i = 0; i < 2; ++i) {
        // H = (feature @ gcn_W[i]) * d_inv_sqrt[row]
        gemm_wmma_kernel<<<ggrid, 256, 0, stream>>>(
            feat, gcnW + (size_t)i * F_DIM * F_DIM, H, nullptr, dinv, nullptr, NN, 1);
        // Z = gather/scatter aggregation over edges
        zero_kernel<<<(tot + 255) / 256, 256, 0, stream>>>(Z, tot);
        scatter_kernel<<<sc_blocks, 256, 0, stream>>>(ei, ewp, H, Z, E);
        // feature = relu(d_inv_sqrt[r] * Z + gcn_b[i])
        gcn_post_kernel<<<(tot + 255) / 256, 256, 0, stream>>>(
            Z, dinv, gcnB + (size_t)i * F_DIM, feat);
        // feature_sum += leaky_relu(feature @ fc_W[i] + fc_b[i])
        gemm_wmma_kernel<<<ggrid, 256, 0, stream>>>(
            feat, fcW + (size_t)i * F_DIM * F_DIM, nullptr, out, nullptr,
            fcB + (size_t)i * F_DIM, NN, 2);
    }
}